// PtModule_76166950028047
// MI455X (gfx1250) — compile-verified
//
#include <hip/hip_runtime.h>

// y = x*x + 2*x == x*(x+2), elementwise over 64Mi fp32 (8192x8192).
// HBM-bound: 512 MB total traffic -> ~22us floor at 23.3 TB/s.
// Exact-path kernel: branchless, 4x global_load_b128 (NT) with immediate
// offsets from one base address, then 4x global_store_b128 (NT).

typedef float v4f __attribute__((ext_vector_type(4)));

#define THREADS 256
#define VEC_PER_THREAD 4
#define VEC_PER_BLOCK (THREADS * VEC_PER_THREAD)  // 1024 float4 per block

// Aligned case: n4 % VEC_PER_BLOCK == 0. No loop, no guards.
__global__ __launch_bounds__(THREADS) void sq_plus_2x_exact(
    const v4f* __restrict__ x, v4f* __restrict__ y) {
  // Thread t of block b handles vec4 indices base+{0,256,512,768}*? -> use
  // consecutive-within-block layout: base = b*1024 + t, offsets k*256 vec4
  // = k*4096 bytes, which fit the signed 24-bit IOFFSET field directly.
  const long long base = (long long)blockIdx.x * VEC_PER_BLOCK + threadIdx.x;
  const v4f* __restrict__ xp = x + base;
  v4f* __restrict__ yp = y + base;

  v4f a0 = __builtin_nontemporal_load(xp + 0 * THREADS);
  v4f a1 = __builtin_nontemporal_load(xp + 1 * THREADS);
  v4f a2 = __builtin_nontemporal_load(xp + 2 * THREADS);
  v4f a3 = __builtin_nontemporal_load(xp + 3 * THREADS);

  v4f r0 = a0 * (a0 + 2.0f);
  v4f r1 = a1 * (a1 + 2.0f);
  v4f r2 = a2 * (a2 + 2.0f);
  v4f r3 = a3 * (a3 + 2.0f);

  __builtin_nontemporal_store(r0, yp + 0 * THREADS);
  __builtin_nontemporal_store(r1, yp + 1 * THREADS);
  __builtin_nontemporal_store(r2, yp + 2 * THREADS);
  __builtin_nontemporal_store(r3, yp + 3 * THREADS);
}

// Generic fallback (any n): guarded grid-stride over float4.
__global__ __launch_bounds__(THREADS) void sq_plus_2x_generic(
    const v4f* __restrict__ x, v4f* __restrict__ y, long long n4) {
  const long long stride = (long long)gridDim.x * blockDim.x;
  for (long long i = (long long)blockIdx.x * blockDim.x + threadIdx.x;
       i < n4; i += stride) {
    v4f a = __builtin_nontemporal_load(&x[i]);
    __builtin_nontemporal_store(a * (a + 2.0f), &y[i]);
  }
}

// Scalar tail for n % 4 != 0 (never hit for 8192*8192).
__global__ void sq_plus_2x_tail(const float* __restrict__ x,
                                float* __restrict__ y,
                                long long start, long long n) {
  long long i = start + (long long)blockIdx.x * blockDim.x + threadIdx.x;
  if (i < n) {
    float v = x[i];
    y[i] = v * (v + 2.0f);
  }
}

extern "C" void kernel_launch(void* const* d_in, const int* in_sizes, int n_in,
                              void* d_out, int out_size, void* d_ws, size_t ws_size,
                              hipStream_t stream) {
  const float* x = (const float*)d_in[0];
  float* y = (float*)d_out;
  const long long n  = (long long)in_sizes[0];   // 67,108,864
  const long long n4 = n / 4;                    // 16,777,216 float4
  const long long rem = n - 4 * n4;

  if (n4 > 0) {
    if (n4 % VEC_PER_BLOCK == 0) {
      const long long blocks = n4 / VEC_PER_BLOCK;  // 16384
      sq_plus_2x_exact<<<(int)blocks, THREADS, 0, stream>>>(
          (const v4f*)x, (v4f*)y);
    } else {
      long long blocks = (n4 + THREADS - 1) / THREADS;
      if (blocks > 262144LL) blocks = 262144LL;
      sq_plus_2x_generic<<<(int)blocks, THREADS, 0, stream>>>(
          (const v4f*)x, (v4f*)y, n4);
    }
  }
  if (rem > 0) {
    sq_plus_2x_tail<<<1, 256, 0, stream>>>(x, y, 4 * n4, n);
  }
}